// MemoryPool_82214263980182
// MI455X (gfx1250) — compile-verified
//
#include <hip/hip_runtime.h>
#include <hip/hip_bf16.h>
#include <stdint.h>

// Problem shape (harness-fixed): N=16384 items, D=1024 dims, P=4096 slots.
#define POOL_MAX     4096          // max slots supported by the LDS tree
#define LEAF0        POOL_MAX      // heap: root=1, leaves at [4096, 8192)
#define CHUNK        16384         // score staging chunk (floats) -> 64KB LDS
#define SCAN_THREADS 256           // 8 wave32 waves

// ---------------------------------------------------------------------------
// CDNA5 async global->LDS (ASYNCcnt path). Generic LDS pointer low 32 bits are
// the wave-relative LDS offset (addrspacecast flat->local == truncate), which
// is exactly what the VDST operand of global_load_async_to_lds wants.
// ---------------------------------------------------------------------------
__device__ __forceinline__ void async_load_b128_to_lds(void* lds_dst, const void* gsrc) {
  unsigned lds_off = (unsigned)(size_t)lds_dst;
  asm volatile("global_load_async_to_lds_b128 %0, %1, off"
               :: "v"(lds_off), "v"(gsrc)
               : "memory");
}

__device__ __forceinline__ void wait_asynccnt0() {
#if __has_builtin(__builtin_amdgcn_s_wait_asynccnt)
  __builtin_amdgcn_s_wait_asynccnt(0);
#else
  asm volatile("s_wait_asynccnt 0" ::: "memory");
#endif
}

// ---------------------------------------------------------------------------
// Kernel A: sequential decision scan with an LDS tournament min-tree.
//   key(slot) = (bits(priority) << 32) | slot     (priorities >= 0, so
//   unsigned compare == float compare, and min-of-keys == jnp.argmin with
//   first-index tie-break).
// Emits: lastWriter[slot] (global ws), final priorities, final count.
// ---------------------------------------------------------------------------
__global__ __launch_bounds__(SCAN_THREADS)
void pool_scan_kernel(const float* __restrict__ scores,
                      const float* __restrict__ priorities,
                      const int*   __restrict__ count0,
                      int*         __restrict__ lastWriter,
                      float*       __restrict__ pri_out,
                      float*       __restrict__ cnt_out,
                      int N, int P) {
  __shared__ __align__(16) unsigned long long tree[2 * POOL_MAX]; // 64 KB
  __shared__ __align__(16) float s_scores[CHUNK];                 // 64 KB

  const int tid  = threadIdx.x;
  const int bdim = blockDim.x;

  // --- cooperative: build leaves + init lastWriter --------------------------
  for (int s = tid; s < POOL_MAX; s += bdim) {
    unsigned long long key = ~0ull;                // padding: never the min
    if (s < P)
      key = ((unsigned long long)__float_as_uint(priorities[s]) << 32) | (unsigned)s;
    tree[LEAF0 + s] = key;
  }
  for (int s = tid; s < P; s += bdim) lastWriter[s] = -1;
  __syncthreads();

  // --- cooperative: build internal min levels bottom-up ---------------------
  for (int lvl = POOL_MAX >> 1; lvl >= 1; lvl >>= 1) {
    for (int n = lvl + tid; n < 2 * lvl; n += bdim) {
      unsigned long long a = tree[2 * n], b = tree[2 * n + 1];
      tree[n] = (a < b) ? a : b;
    }
    __syncthreads();
  }

  unsigned cnt = 0;
  if (tid == 0) cnt = (unsigned)*count0;

  // --- chunked scan: async-stage scores, then thread 0 runs the serial part -
  for (int base = 0; base < N; base += CHUNK) {
    int nflt = N - base; if (nflt > CHUNK) nflt = CHUNK;
    int nvec = nflt >> 2;
    for (int v = tid; v < nvec; v += bdim)
      async_load_b128_to_lds(&s_scores[v * 4], scores + base + v * 4);
    for (int j = (nvec << 2) + tid; j < nflt; j += bdim)  // scalar tail
      s_scores[j] = scores[base + j];
    wait_asynccnt0();
    __syncthreads();

    if (tid == 0) {
      for (int i = 0; i < nflt; ++i) {
        float s = s_scores[i];
        if (!(s > 0.5f)) continue;                 // valid = score > THRESHOLD
        unsigned idx;
        if (cnt < (unsigned)P) {
          idx = cnt++;                             // append while room
        } else {
          unsigned long long root = tree[1];       // O(1) argmin
          float mv = __uint_as_float((unsigned)(root >> 32));
          if (!(s > mv)) continue;                 // replace_ok
          idx = (unsigned)root;                    // low 32 bits = slot
        }
        // update leaf, propagate with early exit (fill phase ~1 level)
        unsigned node = LEAF0 + idx;
        tree[node] = ((unsigned long long)__float_as_uint(s) << 32) | idx;
        node >>= 1;
        while (node) {
          unsigned long long a = tree[2 * node], b = tree[2 * node + 1];
          unsigned long long m = (a < b) ? a : b;
          if (tree[node] == m) break;
          tree[node] = m;
          node >>= 1;
        }
        lastWriter[idx] = base + i;                // last write wins
      }
    }
    __syncthreads();
  }

  // --- outputs --------------------------------------------------------------
  if (tid == 0) *cnt_out = (float)cnt;
  for (int s = tid; s < P; s += bdim)
    pri_out[s] = __uint_as_float((unsigned)(tree[LEAF0 + s] >> 32));
}

// ---------------------------------------------------------------------------
// Kernel B: bandwidth-bound gather. One block per slot; copy the last-writer
// summary row (or the untouched original pool row). 32 MB total traffic ->
// ~1.4 us at 23.3 TB/s; working set is L2-resident (192 MB).
// ---------------------------------------------------------------------------
__global__ __launch_bounds__(256)
void pool_scatter_kernel(const float* __restrict__ summaries,
                         const float* __restrict__ pool_in,
                         const int*   __restrict__ lastWriter,
                         float*       __restrict__ pool_out,
                         int D) {
  const int s = blockIdx.x;
  const int w = lastWriter[s];
  const float* src = (w >= 0) ? summaries + (size_t)w * D
                              : pool_in   + (size_t)s * D;
  float* dst = pool_out + (size_t)s * D;

  const int nv = D >> 2;
  const float4* s4 = (const float4*)src;
  float4*       d4 = (float4*)dst;
  for (int j = threadIdx.x; j < nv; j += blockDim.x) d4[j] = s4[j];
  for (int j = (nv << 2) + threadIdx.x; j < D; j += blockDim.x) dst[j] = src[j];
}

// ---------------------------------------------------------------------------
extern "C" void kernel_launch(void* const* d_in, const int* in_sizes, int n_in,
                              void* d_out, int out_size, void* d_ws, size_t ws_size,
                              hipStream_t stream) {
  const float* summaries  = (const float*)d_in[0];   // (N, D) f32
  const float* scores     = (const float*)d_in[1];   // (N,)   f32
  const float* pool       = (const float*)d_in[2];   // (P, D) f32
  const float* priorities = (const float*)d_in[3];   // (P,)   f32
  const int*   count      = (const int*)  d_in[4];   // scalar int

  const int N = in_sizes[1];
  const int P = in_sizes[3];
  const int D = (N > 0) ? (in_sizes[0] / N) : 0;

  float* out      = (float*)d_out;
  float* pool_out = out;                       // P*D floats
  float* pri_out  = out + (size_t)P * D;       // P floats
  float* cnt_out  = pri_out + P;               // 1 float (numeric count)

  int* lastWriter = (int*)d_ws;                // P ints of scratch

  pool_scan_kernel<<<1, SCAN_THREADS, 0, stream>>>(
      scores, priorities, count, lastWriter, pri_out, cnt_out, N, P);

  pool_scatter_kernel<<<P, 256, 0, stream>>>(
      summaries, pool, lastWriter, pool_out, D);
}